// TemporalGCN_65635690218230
// MI455X (gfx1250) — compile-verified
//
#include <hip/hip_runtime.h>
#include <math.h>

// CDNA5 / gfx1250, wave32. fp32 WMMA 16x16x4: A,B = v2f (2 VGPR), C/D = v8f (8 VGPR).
typedef float v2f __attribute__((ext_vector_type(2)));
typedef float v8f __attribute__((ext_vector_type(8)));

#define WMMA_F32(a, b, c) \
  __builtin_amdgcn_wmma_f32_16x16x4_f32(false, (a), false, (b), (short)0, (c), false, false)

// CDNA5 async global->LDS staging (ASYNCcnt path); falls back to sync float4 copies.
#if __has_builtin(__builtin_amdgcn_global_load_async_to_lds_b128) && \
    __has_builtin(__builtin_amdgcn_s_wait_asynccnt)
#define USE_ASYNC_LDS 1
#endif

#ifdef USE_ASYNC_LDS
typedef int v4i_vs __attribute__((__vector_size__(16)));
#endif

// copy 16 bytes global -> LDS (per calling thread)
static __device__ __forceinline__ void copy16_to_lds(float* dst_lds, const float* src_global) {
#ifdef USE_ASYNC_LDS
  // builtin signature (per hipcc diagnostic): (v4i AS(1)*, v4i AS(3)*, imm offset, imm cpol)
  __attribute__((address_space(1))) v4i_vs* src =
      reinterpret_cast<__attribute__((address_space(1))) v4i_vs*>(
          reinterpret_cast<uintptr_t>(src_global));
  // generic LDS pointer: low 32 bits are the LDS byte offset
  __attribute__((address_space(3))) v4i_vs* dst =
      reinterpret_cast<__attribute__((address_space(3))) v4i_vs*>(
          static_cast<unsigned int>(reinterpret_cast<uintptr_t>(dst_lds)));
  __builtin_amdgcn_global_load_async_to_lds_b128(src, dst, 0, 0);
#else
  *(float4*)dst_lds = *(const float4*)src_global;
#endif
}

static __device__ __forceinline__ void lds_copy_wait() {
#ifdef USE_ASYNC_LDS
  __builtin_amdgcn_s_wait_asynccnt(0);
#endif
}

// sum across a 16-lane half (lanes 0-15 or 16-31); xor masks <=8 stay in-half
static __device__ __forceinline__ float half_reduce(float v) {
  v += __shfl_xor(v, 1, 32);
  v += __shfl_xor(v, 2, 32);
  v += __shfl_xor(v, 4, 32);
  v += __shfl_xor(v, 8, 32);
  return v;
}

// ---------------- CSR build ----------------
__global__ void k_init_counts(int* counts, int n) {
  int i = blockIdx.x * blockDim.x + threadIdx.x;
  if (i < n) counts[i] = 0;
}

__global__ void k_count(const long long* __restrict__ ei, int E, int* counts) {
  int e = blockIdx.x * blockDim.x + threadIdx.x;
  if (e < E) atomicAdd(&counts[(int)ei[E + e]], 1);
}

__global__ void k_dinv(const int* __restrict__ counts, float* __restrict__ dinv, int n) {
  int i = blockIdx.x * blockDim.x + threadIdx.x;
  if (i < n) dinv[i] = rsqrtf(1.0f + (float)counts[i]);  // +1 self loop
}

// single-block exclusive scan of counts -> rowstart/fill; rowstart[n]=E
__global__ __launch_bounds__(1024) void k_scan(const int* __restrict__ counts,
                                               int* __restrict__ rowstart,
                                               int* __restrict__ fill, int n) {
  __shared__ int smem[1024];
  __shared__ int carry;
  int t = threadIdx.x;
  if (t == 0) carry = 0;
  __syncthreads();
  for (int base = 0; base < n; base += 1024) {
    int i = base + t;
    int v = (i < n) ? counts[i] : 0;
    smem[t] = v;
    __syncthreads();
    for (int off = 1; off < 1024; off <<= 1) {
      int tmp = (t >= off) ? smem[t - off] : 0;
      __syncthreads();
      smem[t] += tmp;
      __syncthreads();
    }
    int excl = smem[t] - v + carry;
    if (i < n) { rowstart[i] = excl; fill[i] = excl; }
    __syncthreads();
    if (t == 0) carry += smem[1023];
    __syncthreads();
  }
  if (t == 0) rowstart[n] = carry;
}

__global__ void k_fill(const long long* __restrict__ ei, int E,
                       int* __restrict__ fill, int* __restrict__ csr_src) {
  int e = blockIdx.x * blockDim.x + threadIdx.x;
  if (e < E) {
    int d = (int)ei[E + e];
    int pos = atomicAdd(&fill[d], 1);
    csr_src[pos] = (int)ei[e];
  }
}

// ---------------- xe = relu(x @ Wn + bn) ----------------
// block = 256 (8 waves): one 16-row M tile, wave w -> N-tile w (16 cols)
__global__ __launch_bounds__(256) void k_gemm_relu(const float* __restrict__ A,
                                                   const float* __restrict__ W,
                                                   const float* __restrict__ bias,
                                                   float* __restrict__ out) {
  __shared__ float As[16 * 128];
  const int t = threadIdx.x;
  const int base = blockIdx.x * 16;
  const float* src = A + (size_t)base * 128;
  copy16_to_lds(&As[t * 4], src + t * 4);
  copy16_to_lds(&As[1024 + t * 4], src + 1024 + t * 4);
  lds_copy_wait();
  __syncthreads();
  const int w = t >> 5, lane = t & 31;
  const int col = w * 16 + (lane & 15), r = lane & 15, kh = (lane < 16) ? 0 : 2;
  v8f c = {};
#pragma unroll 4
  for (int kk = 0; kk < 128; kk += 4) {
    v2f a, b;
    a.x = As[r * 128 + kk + kh];
    a.y = As[r * 128 + kk + kh + 1];
    b.x = W[(kk + kh) * 128 + col];
    b.y = W[(kk + kh + 1) * 128 + col];
    c = WMMA_F32(a, b, c);
  }
  const float bb = bias[col];
#pragma unroll
  for (int v = 0; v < 8; v++) {
    int row = v + ((lane < 16) ? 0 : 8);
    out[(size_t)(base + row) * 128 + col] = fmaxf(c[v] + bb, 0.0f);
  }
}

// ---------------- m[n] = dinv[n]*sum_{e:dst=n} dinv[src]*xe[src] + dinv[n]^2*xe[n] ----------------
__global__ __launch_bounds__(256) void k_aggregate(const float* __restrict__ xe,
                                                   const float* __restrict__ dinv,
                                                   const int* __restrict__ rowstart,
                                                   const int* __restrict__ csr_src,
                                                   float* __restrict__ m, int n) {
  int node = (blockIdx.x * blockDim.x + threadIdx.x) >> 5;  // one wave32 per node
  int lane = threadIdx.x & 31;
  if (node >= n) return;
  float4 acc = {0.f, 0.f, 0.f, 0.f};
  int s0 = rowstart[node], s1 = rowstart[node + 1];
  for (int i = s0; i < s1; i++) {
    int s = csr_src[i];
    float ds = dinv[s];
    float4 v = *(const float4*)(xe + (size_t)s * 128 + lane * 4);
    acc.x += ds * v.x; acc.y += ds * v.y; acc.z += ds * v.z; acc.w += ds * v.w;
  }
  float dn = dinv[node], dn2 = dn * dn;
  float4 sf = *(const float4*)(xe + (size_t)node * 128 + lane * 4);
  acc.x = dn * acc.x + dn2 * sf.x;
  acc.y = dn * acc.y + dn2 * sf.y;
  acc.z = dn * acc.z + dn2 * sf.z;
  acc.w = dn * acc.w + dn2 * sf.w;
  *(float4*)(m + (size_t)node * 128 + lane * 4) = acc;
}

// ---------------- gz = m@Wzc+bzc ; gh = m@Whc+bhc (shared A tile) ----------------
__global__ __launch_bounds__(256) void k_gemm_dual(const float* __restrict__ A,
                                                   const float* __restrict__ W1,
                                                   const float* __restrict__ b1,
                                                   float* __restrict__ o1,
                                                   const float* __restrict__ W2,
                                                   const float* __restrict__ b2,
                                                   float* __restrict__ o2) {
  __shared__ float As[16 * 128];
  const int t = threadIdx.x;
  const int base = blockIdx.x * 16;
  const float* src = A + (size_t)base * 128;
  copy16_to_lds(&As[t * 4], src + t * 4);
  copy16_to_lds(&As[1024 + t * 4], src + 1024 + t * 4);
  lds_copy_wait();
  __syncthreads();
  const int w = t >> 5, lane = t & 31;
  const int col = w * 16 + (lane & 15), r = lane & 15, kh = (lane < 16) ? 0 : 2;
  v8f c1 = {}, c2 = {};
#pragma unroll 4
  for (int kk = 0; kk < 128; kk += 4) {
    v2f a, b1f, b2f;
    a.x = As[r * 128 + kk + kh];
    a.y = As[r * 128 + kk + kh + 1];
    b1f.x = W1[(kk + kh) * 128 + col];
    b1f.y = W1[(kk + kh + 1) * 128 + col];
    b2f.x = W2[(kk + kh) * 128 + col];
    b2f.y = W2[(kk + kh + 1) * 128 + col];
    c1 = WMMA_F32(a, b1f, c1);
    c2 = WMMA_F32(a, b2f, c2);
  }
  const float bb1 = b1[col], bb2 = b2[col];
#pragma unroll
  for (int v = 0; v < 8; v++) {
    int row = v + ((lane < 16) ? 0 : 8);
    o1[(size_t)(base + row) * 128 + col] = c1[v] + bb1;
    o2[(size_t)(base + row) * 128 + col] = c2[v] + bb2;
  }
}

// ---------------- H = (1-sigmoid(gz@Wzl1+bzl)) * tanh(gh@Whl1+bhl);  aS=H.w1, aD=H.w2 ----------------
__global__ __launch_bounds__(256) void k_gru_head(const float* __restrict__ Gz,
                                                  const float* __restrict__ Gh,
                                                  const float* __restrict__ Wzl,
                                                  const float* __restrict__ bzl,
                                                  const float* __restrict__ Whl,
                                                  const float* __restrict__ bhl,
                                                  const float* __restrict__ Wout,
                                                  float* __restrict__ aS,
                                                  float* __restrict__ aD) {
  __shared__ float Zs[16 * 128];
  __shared__ float Hs[16 * 128];
  __shared__ float sA[16], sB[16];
  const int t = threadIdx.x;
  const int base = blockIdx.x * 16;
  const float* gz = Gz + (size_t)base * 128;
  const float* gh = Gh + (size_t)base * 128;
  copy16_to_lds(&Zs[t * 4], gz + t * 4);
  copy16_to_lds(&Zs[1024 + t * 4], gz + 1024 + t * 4);
  copy16_to_lds(&Hs[t * 4], gh + t * 4);
  copy16_to_lds(&Hs[1024 + t * 4], gh + 1024 + t * 4);
  if (t < 16) { sA[t] = 0.f; sB[t] = 0.f; }
  lds_copy_wait();
  __syncthreads();
  const int w = t >> 5, lane = t & 31;
  const int col = w * 16 + (lane & 15), r = lane & 15, kh = (lane < 16) ? 0 : 2;
  v8f cz = {}, ch = {};
#pragma unroll 4
  for (int kk = 0; kk < 128; kk += 4) {
    v2f az, ah, bz, bh;
    az.x = Zs[r * 128 + kk + kh];
    az.y = Zs[r * 128 + kk + kh + 1];
    ah.x = Hs[r * 128 + kk + kh];
    ah.y = Hs[r * 128 + kk + kh + 1];
    // Wzl/Whl are [256,128]; H==0 kills rows 128..255, so only the first 128 rows matter
    bz.x = Wzl[(kk + kh) * 128 + col];
    bz.y = Wzl[(kk + kh + 1) * 128 + col];
    bh.x = Whl[(kk + kh) * 128 + col];
    bh.y = Whl[(kk + kh + 1) * 128 + col];
    cz = WMMA_F32(az, bz, cz);
    ch = WMMA_F32(ah, bh, ch);
  }
  const float bbz = bzl[col], bbh = bhl[col];
  const float w1 = Wout[col], w2 = Wout[128 + col];
  float s1[8], s2[8];
#pragma unroll
  for (int v = 0; v < 8; v++) {
    float z = 1.0f / (1.0f + expf(-(cz[v] + bbz)));
    float h = tanhf(ch[v] + bbh);
    float Hv = (1.0f - z) * h;
    s1[v] = half_reduce(Hv * w1);
    s2[v] = half_reduce(Hv * w2);
  }
  if (lane == 0) {
#pragma unroll
    for (int v = 0; v < 8; v++) { atomicAdd(&sA[v], s1[v]); atomicAdd(&sB[v], s2[v]); }
  } else if (lane == 16) {
#pragma unroll
    for (int v = 0; v < 8; v++) { atomicAdd(&sA[8 + v], s1[v]); atomicAdd(&sB[8 + v], s2[v]); }
  }
  __syncthreads();
  if (t < 16) { aS[base + t] = sA[t]; aD[base + t] = sB[t]; }
}

// ---------------- out[e] = relu(edge_attr[e]@We+be).w3 + aS[src]+aD[dst]+bout ----------------
__global__ __launch_bounds__(256) void k_edge_out(const float* __restrict__ EA,
                                                  const float* __restrict__ We,
                                                  const float* __restrict__ be,
                                                  const float* __restrict__ Wout,
                                                  const float* __restrict__ bout,
                                                  const long long* __restrict__ ei, int E,
                                                  const float* __restrict__ aS,
                                                  const float* __restrict__ aD,
                                                  float* __restrict__ out) {
  __shared__ float As[16 * 64];
  __shared__ float tsum[16];
  const int t = threadIdx.x;
  const int base = blockIdx.x * 16;
  copy16_to_lds(&As[t * 4], EA + (size_t)base * 64 + t * 4);  // 1024 floats
  if (t < 16) tsum[t] = 0.f;
  lds_copy_wait();
  __syncthreads();
  const int w = t >> 5, lane = t & 31;
  const int col = w * 16 + (lane & 15), r = lane & 15, kh = (lane < 16) ? 0 : 2;
  v8f c = {};
#pragma unroll
  for (int kk = 0; kk < 64; kk += 4) {
    v2f a, b;
    a.x = As[r * 64 + kk + kh];
    a.y = As[r * 64 + kk + kh + 1];
    b.x = We[(kk + kh) * 128 + col];
    b.y = We[(kk + kh + 1) * 128 + col];
    c = WMMA_F32(a, b, c);
  }
  const float bb = be[col], w3 = Wout[256 + col];
  float s[8];
#pragma unroll
  for (int v = 0; v < 8; v++) s[v] = half_reduce(fmaxf(c[v] + bb, 0.f) * w3);
  if (lane == 0) {
#pragma unroll
    for (int v = 0; v < 8; v++) atomicAdd(&tsum[v], s[v]);
  } else if (lane == 16) {
#pragma unroll
    for (int v = 0; v < 8; v++) atomicAdd(&tsum[8 + v], s[v]);
  }
  __syncthreads();
  if (t < 16) {
    int e = base + t;
    int sc = (int)ei[e], dc = (int)ei[E + e];
    out[e] = tsum[t] + aS[sc] + aD[dc] + bout[0];
  }
}

extern "C" void kernel_launch(void* const* d_in, const int* in_sizes, int n_in,
                              void* d_out, int out_size, void* d_ws, size_t ws_size,
                              hipStream_t stream) {
  const float* x = (const float*)d_in[0];
  const long long* ei = (const long long*)d_in[1];
  const float* ea = (const float*)d_in[2];
  const float* Wn = (const float*)d_in[3];
  const float* bn = (const float*)d_in[4];
  const float* We = (const float*)d_in[5];
  const float* be = (const float*)d_in[6];
  const float* Wzc = (const float*)d_in[7];
  const float* bzc = (const float*)d_in[8];
  const float* Wzl = (const float*)d_in[9];
  const float* bzl = (const float*)d_in[10];
  // d_in[11..14] = Wrc,brc,Wrl,brl: dead (H==0 -> H*R==0)
  const float* Whc = (const float*)d_in[15];
  const float* bhc = (const float*)d_in[16];
  const float* Whl = (const float*)d_in[17];
  const float* bhl = (const float*)d_in[18];
  const float* Wout = (const float*)d_in[19];
  const float* bout = (const float*)d_in[20];
  float* out = (float*)d_out;

  const int N = in_sizes[0] / 128;  // 50000
  const int E = in_sizes[2] / 64;   // 800000

  // workspace layout (floats then ints, all 4B aligned)
  char* p = (char*)d_ws;
  float* xe_gz = (float*)p; p += (size_t)N * 128 * 4;  // xe, reused as gz
  float* m     = (float*)p; p += (size_t)N * 128 * 4;
  float* gh    = (float*)p; p += (size_t)N * 128 * 4;
  float* dinv  = (float*)p; p += (size_t)N * 4;
  float* aSbuf = (float*)p; p += (size_t)N * 4;
  float* aDbuf = (float*)p; p += (size_t)N * 4;
  int* counts   = (int*)p; p += (size_t)N * 4;
  int* rowstart = (int*)p; p += (size_t)(N + 1) * 4;
  int* fill     = (int*)p; p += (size_t)N * 4;
  int* csr_src  = (int*)p; p += (size_t)E * 4;

  // CSR build + normalization
  k_init_counts<<<(N + 255) / 256, 256, 0, stream>>>(counts, N);
  k_count<<<(E + 255) / 256, 256, 0, stream>>>(ei, E, counts);
  k_dinv<<<(N + 255) / 256, 256, 0, stream>>>(counts, dinv, N);
  k_scan<<<1, 1024, 0, stream>>>(counts, rowstart, fill, N);
  k_fill<<<(E + 255) / 256, 256, 0, stream>>>(ei, E, fill, csr_src);

  // node pipeline (N divisible by 16)
  k_gemm_relu<<<N / 16, 256, 0, stream>>>(x, Wn, bn, xe_gz);              // xe
  k_aggregate<<<(N + 7) / 8, 256, 0, stream>>>(xe_gz, dinv, rowstart, csr_src, m, N);
  k_gemm_dual<<<N / 16, 256, 0, stream>>>(m, Wzc, bzc, xe_gz /*gz*/, Whc, bhc, gh);
  k_gru_head<<<N / 16, 256, 0, stream>>>(xe_gz, gh, Wzl, bzl, Whl, bhl, Wout, aSbuf, aDbuf);

  // edge pipeline (E divisible by 16)
  k_edge_out<<<E / 16, 256, 0, stream>>>(ea, We, be, Wout, bout, ei, E, aSbuf, aDbuf, out);
}